// Warp_8315056685089
// MI455X (gfx1250) — compile-verified
//
#include <hip/hip_runtime.h>

// Problem constants from the reference: B,C,H,W = 8,64,256,256
#define Bc 8
#define Cc 64
#define Hc 256
#define Wc 256
#define HWc (Hc * Wc)   // 65536
#define BLK 256

__global__ __launch_bounds__(BLK) void warp_bilinear_kernel(
    const float* __restrict__ phi,    // [B,2,H,W]
    const float* __restrict__ x_enc,  // [B,C,H,W]
    const float* __restrict__ msk,    // [B,1,H,W]
    float* __restrict__ out)          // [B,C,H,W]
{
    __shared__ float s_p0[BLK];
    __shared__ float s_p1[BLK];
    __shared__ float s_m[BLK];

    const int tid = threadIdx.x;
    const int pix = blockIdx.x * BLK + tid;   // 0 .. B*H*W-1
    const int b   = pix >> 16;                // / HW
    const int hw  = pix & (HWc - 1);
    const int y   = hw >> 8;
    const int x   = hw & (Wc - 1);

    // ---- CDNA5 async global->LDS staging of phi.x / phi.y / m (coalesced B32,
    //      tracked with ASYNCcnt; waited on before the index math below) ----
    {
        const float* g0 = phi + (size_t)(b * 2 + 0) * HWc + hw;
        const float* g1 = phi + (size_t)(b * 2 + 1) * HWc + hw;
        const float* g2 = msk + (size_t)b * HWc + hw;
        unsigned l0 = (unsigned)(size_t)&s_p0[tid];
        unsigned l1 = (unsigned)(size_t)&s_p1[tid];
        unsigned l2 = (unsigned)(size_t)&s_m[tid];
        asm volatile("global_load_async_to_lds_b32 %0, %1, off"
                     :: "v"(l0), "v"(g0) : "memory");
        asm volatile("global_load_async_to_lds_b32 %0, %1, off"
                     :: "v"(l1), "v"(g1) : "memory");
        asm volatile("global_load_async_to_lds_b32 %0, %1, off"
                     :: "v"(l2), "v"(g2) : "memory");
    }

    // Each lane only reads the LDS words it wrote itself -> ASYNCcnt wait is
    // sufficient, no workgroup barrier required.
    asm volatile("s_wait_asynccnt 0" ::: "memory");

    const float p0 = s_p0[tid];
    const float p1 = s_p1[tid];
    const float mv = s_m[tid];

    // vgrid = grid + phi; normalize; then += 2*phi (reference quirk)
    const float gx = 2.0f * ((float)x + p0) * (1.0f / (float)(Wc - 1)) - 1.0f + 2.0f * p0;
    const float gy = 2.0f * ((float)y + p1) * (1.0f / (float)(Hc - 1)) - 1.0f + 2.0f * p1;

    // grid_sample, align_corners=False
    const float ix = (gx + 1.0f) * (0.5f * (float)Wc) - 0.5f;
    const float iy = (gy + 1.0f) * (0.5f * (float)Hc) - 0.5f;

    const float x0f = floorf(ix);
    const float y0f = floorf(iy);
    const int   x0  = (int)x0f;
    const int   y0  = (int)y0f;
    const int   x1  = x0 + 1;
    const int   y1  = y0 + 1;

    const float wx1 = ix - x0f, wx0 = 1.0f - wx1;
    const float wy1 = iy - y0f, wy0 = 1.0f - wy1;

    const bool vx0 = (x0 >= 0) & (x0 < Wc);
    const bool vx1 = (x1 >= 0) & (x1 < Wc);
    const bool vy0 = (y0 >= 0) & (y0 < Hc);
    const bool vy1 = (y1 >= 0) & (y1 < Hc);

    const int xc0 = min(max(x0, 0), Wc - 1);
    const int xc1 = min(max(x1, 0), Wc - 1);
    const int yc0 = min(max(y0, 0), Hc - 1);
    const int yc1 = min(max(y1, 0), Hc - 1);

    // Fold zero-padding validity AND the mask m into the 4 corner weights:
    // the channel loop becomes 4 gathers + 4 FMAs + 1 NT store.
    const float w00 = wy0 * wx0 * (float)(vy0 & vx0) * mv;
    const float w01 = wy0 * wx1 * (float)(vy0 & vx1) * mv;
    const float w10 = wy1 * wx0 * (float)(vy1 & vx0) * mv;
    const float w11 = wy1 * wx1 * (float)(vy1 & vx1) * mv;

    const int o00 = yc0 * Wc + xc0;
    const int o01 = yc0 * Wc + xc1;
    const int o10 = yc1 * Wc + xc0;
    const int o11 = yc1 * Wc + xc1;

    const float* __restrict__ src = x_enc + (size_t)b * Cc * HWc;
    float* __restrict__ dst = out + (size_t)b * Cc * HWc + hw;

    // x_enc (134 MB) fits in the 192 MB L2 -> keep its loads regular-temporal.
    // out (134 MB) is streamed exactly once -> non-temporal stores so the
    // output stream does not evict x_enc from L2.
#pragma unroll 8
    for (int c = 0; c < Cc; ++c) {
        const float* __restrict__ p = src + c * HWc;
        float v00 = p[o00];
        float v01 = p[o01];
        float v10 = p[o10];
        float v11 = p[o11];
        float v = fmaf(v00, w00, fmaf(v01, w01, fmaf(v10, w10, v11 * w11)));
        __builtin_nontemporal_store(v, dst + c * HWc);
    }
}

extern "C" void kernel_launch(void* const* d_in, const int* in_sizes, int n_in,
                              void* d_out, int out_size, void* d_ws, size_t ws_size,
                              hipStream_t stream) {
    const float* phi = (const float*)d_in[0];  // [8,2,256,256]
    const float* xe  = (const float*)d_in[1];  // [8,64,256,256]
    const float* m   = (const float*)d_in[2];  // [8,1,256,256]
    float* out = (float*)d_out;                // [8,64,256,256]

    const int total_pix = Bc * HWc;            // 524288
    dim3 grid(total_pix / BLK);                // 2048 blocks
    dim3 block(BLK);                           // 8 wave32s
    warp_bilinear_kernel<<<grid, block, 0, stream>>>(phi, xe, m, out);
}